// GAT_86242943303863
// MI455X (gfx1250) — compile-verified
//
#include <hip/hip_runtime.h>
#include <hip/hip_bf16.h>

// Problem constants (match reference)
#define NN   50000
#define EE   800000
#define CH   128            // per-head channels (HID)
#define NEG_SLOPE 0.2f

typedef __attribute__((ext_vector_type(2))) float v2f;
typedef __attribute__((ext_vector_type(8))) float v8f;

__device__ __forceinline__ v8f wmma4(v2f a, v2f b, v8f c) {
    return __builtin_amdgcn_wmma_f32_16x16x4_f32(
               false, a, false, b, (short)0, c, false, false);
}

// ---------- monotone float <-> uint map for atomicMax on signed floats ----------
__device__ __forceinline__ unsigned enc_f32(float f) {
    unsigned u = __float_as_uint(f);
    return (u >> 31) ? ~u : (u | 0x80000000u);
}
__device__ __forceinline__ float dec_f32(unsigned u) {
    return __uint_as_float((u >> 31) ? (u ^ 0x80000000u) : ~u);
}

// ---------- edge decode with virtual self-loops appended ----------
__device__ __forceinline__ void edge_sd(const int* __restrict__ ei, int e, int& s, int& d) {
    if (e < EE) { s = ei[e]; d = ei[EE + e]; }
    else        { s = e - EE; d = e - EE; }
}

// =====================================================================
// GEMM: C[M,Nout] = A[M,K] @ W[Nout,K]^T  (+bias)(+relu), via fp32 WMMA.
// One wave -> one 16x64 strip (4 independent accumulators): the A
// fragment is loaded once and fed to 4 WMMAs per K-step (1.25 VMEM ops
// per WMMA instead of 2). Wave->(m_sub, n_grp) mapping adapts to Nout so
// both 512-wide and 128-wide layers keep all 8 waves busy.
// flags: bit0 = add bias, bit1 = relu
// =====================================================================
__global__ __launch_bounds__(256)
void gemm_wmma_f32(const float* __restrict__ A, const float* __restrict__ W,
                   const float* __restrict__ bias, float* __restrict__ C,
                   int M16 /* #16-row tiles */, int K, int Nout, int flags)
{
    const int wave  = threadIdx.x >> 5;
    const int lane  = threadIdx.x & 31;
    const int mlane = lane & 15;          // M (for A) / N (for B) index inside tile
    const int khi   = lane >> 4;          // selects K pair {0,1} vs {2,3}

    const int ngrps = Nout >> 6;                  // 64-wide column groups
    const int wpn   = (ngrps < 8) ? ngrps : 8;    // waves along N per block
    const int mpb   = 8 / wpn;                    // m-subtiles per block

    const int mt = blockIdx.y * mpb + (wave / wpn);
    if (mt >= M16) return;                        // wave-uniform exit
    const int n0 = (blockIdx.x * wpn + (wave % wpn)) * 64;
    if (n0 >= Nout) return;                       // wave-uniform exit
    const int m0 = mt * 16;

    const float* arow = A + (size_t)(m0 + mlane) * K + 2 * khi;
    const float* wrow = W + (size_t)(n0 + mlane) * K + 2 * khi;
    const size_t wstr = (size_t)16 * K;           // stride between 16-col tiles

    v8f acc0 = {}, acc1 = {}, acc2 = {}, acc3 = {};
#pragma unroll 2
    for (int k0 = 0; k0 < K; k0 += 4) {
        v2f a  = *(const v2f*)(arow + k0);
        v2f b0 = *(const v2f*)(wrow + k0);
        v2f b1 = *(const v2f*)(wrow + wstr + k0);
        v2f b2 = *(const v2f*)(wrow + 2 * wstr + k0);
        v2f b3 = *(const v2f*)(wrow + 3 * wstr + k0);
        acc0 = wmma4(a, b0, acc0);
        acc1 = wmma4(a, b1, acc1);
        acc2 = wmma4(a, b2, acc2);
        acc3 = wmma4(a, b3, acc3);
    }

    v8f accs[4] = {acc0, acc1, acc2, acc3};
#pragma unroll
    for (int j = 0; j < 4; ++j) {
        const int col = n0 + j * 16 + mlane;
        float bv = (flags & 1) ? bias[col] : 0.0f;
#pragma unroll
        for (int r = 0; r < 8; ++r) {
            float v = accs[j][r] + bv;
            if (flags & 2) v = fmaxf(v, 0.0f);
            C[(size_t)(m0 + r + 8 * khi) * Nout + col] = v;
        }
    }
}

// =====================================================================
// alpha_s[n,h] = sum_c h[n,h,c]*a_src[h,c] ; same for alpha_d
// one thread per (node, head)
// =====================================================================
__global__ __launch_bounds__(256)
void alpha_kernel(const float* __restrict__ h, const float* __restrict__ a_src,
                  const float* __restrict__ a_dst, float* __restrict__ as_,
                  float* __restrict__ ad_, int heads)
{
    int idx = blockIdx.x * blockDim.x + threadIdx.x;
    if (idx >= NN * heads) return;
    int n = idx / heads, hh = idx % heads;
    const float* hp  = h + (size_t)n * heads * CH + hh * CH;
    const float* asv = a_src + hh * CH;
    const float* adv = a_dst + hh * CH;
    float s = 0.f, d = 0.f;
#pragma unroll 8
    for (int c = 0; c < CH; ++c) { float v = hp[c]; s += v * asv[c]; d += v * adv[c]; }
    as_[idx] = s; ad_[idx] = d;
}

// =====================================================================
// Pass A over edges: e = leakyrelu(as[src]+ad[dst]); store e; atomicMax m[dst]
// one thread per edge
// =====================================================================
__global__ __launch_bounds__(256)
void edge_max_kernel(const int* __restrict__ ei, const float* __restrict__ as_,
                     const float* __restrict__ ad_, float* __restrict__ e_edge,
                     unsigned* __restrict__ m_, int heads, int Etot)
{
    int e = blockIdx.x * blockDim.x + threadIdx.x;
    if (e >= Etot) return;
    int s, d; edge_sd(ei, e, s, d);
#pragma unroll 4
    for (int h = 0; h < heads; ++h) {
        float v = as_[s * heads + h] + ad_[d * heads + h];
        v = (v > 0.f) ? v : NEG_SLOPE * v;
        e_edge[(size_t)e * heads + h] = v;
        atomicMax(&m_[d * heads + h], enc_f32(v));
    }
}

// =====================================================================
// Pass B over edges: w = exp(e - m[dst]); z[dst] += w (once per head);
// acc[dst,h,c] += w_h * hfeat[src,h,c].   One wave per edge; float4 I/O.
// HC = heads*128 (512 or 128); lane covers 4 channels per i-iteration.
// =====================================================================
__global__ __launch_bounds__(256)
void edge_agg_kernel(const int* __restrict__ ei, const float* __restrict__ e_edge,
                     const unsigned* __restrict__ m_, float* __restrict__ z_,
                     const float* __restrict__ hfeat, float* __restrict__ acc,
                     int heads, int Etot)
{
    int gtid = blockIdx.x * blockDim.x + threadIdx.x;
    int e    = gtid >> 5;
    int lane = gtid & 31;
    if (e >= Etot) return;
    int s, d; edge_sd(ei, e, s, d);

    float w[4];
#pragma unroll 4
    for (int h = 0; h < heads; ++h)
        w[h] = __expf(e_edge[(size_t)e * heads + h] - dec_f32(m_[d * heads + h]));

    if (lane < heads) atomicAdd(&z_[d * heads + lane], w[lane]);

    const int HC = heads * CH;
    const float4* __restrict__ src4 = (const float4*)(hfeat + (size_t)s * HC);
    float* __restrict__ dstp = acc + (size_t)d * HC;
#pragma unroll 4
    for (int i = 0; i < heads; ++i) {          // HC/128 == heads
        int cbase = i * CH + lane * 4;
        float4 v = src4[cbase >> 2];
        float ww = w[i];
        atomicAdd(dstp + cbase + 0, v.x * ww);
        atomicAdd(dstp + cbase + 1, v.y * ww);
        atomicAdd(dstp + cbase + 2, v.z * ww);
        atomicAdd(dstp + cbase + 3, v.w * ww);
    }
}

// =====================================================================
// Pass C over nodes: out = acc / (z + 1e-16) + bias (+relu). In-place OK.
// =====================================================================
__global__ __launch_bounds__(256)
void normalize_kernel(const float* __restrict__ acc, const float* __restrict__ z_,
                      const float* __restrict__ bias, float* __restrict__ outp,
                      int heads, int relu)
{
    int HC = heads * CH;
    long long idx = (long long)blockIdx.x * blockDim.x + threadIdx.x;
    if (idx >= (long long)NN * HC) return;
    int n   = (int)(idx / HC);
    int rem = (int)(idx % HC);
    int h   = rem / CH;
    float v = acc[idx] / (z_[n * heads + h] + 1e-16f) + bias[rem];
    if (relu) v = fmaxf(v, 0.0f);
    outp[idx] = v;
}

// =====================================================================
// Host orchestration
// =====================================================================
extern "C" void kernel_launch(void* const* d_in, const int* in_sizes, int n_in,
                              void* d_out, int out_size, void* d_ws, size_t ws_size,
                              hipStream_t stream)
{
    const float* x      = (const float*)d_in[0];
    const int*   ei     = (const int*)  d_in[1];
    const float* W0     = (const float*)d_in[2];
    const float* a_src0 = (const float*)d_in[3];
    const float* a_dst0 = (const float*)d_in[4];
    const float* b0     = (const float*)d_in[5];
    const float* W1     = (const float*)d_in[6];
    const float* a_src1 = (const float*)d_in[7];
    const float* a_dst1 = (const float*)d_in[8];
    const float* b1     = (const float*)d_in[9];
    const float* W2     = (const float*)d_in[10];
    const float* a_src2 = (const float*)d_in[11];
    const float* a_dst2 = (const float*)d_in[12];
    const float* b2     = (const float*)d_in[13];
    const float* Wv     = (const float*)d_in[14];
    const float* bv     = (const float*)d_in[15];
    const float* Wt     = (const float*)d_in[16];
    const float* bt     = (const float*)d_in[17];
    float* out = (float*)d_out;

    const int Etot = EE + NN;
    const int M16  = NN / 16;   // 3125 16-row tiles

    // workspace carve-up (256B aligned)
    char* ws = (char*)d_ws; size_t off = 0;
    auto carve = [&](size_t bytes) -> void* {
        void* p = ws + off; off = (off + bytes + 255) & ~(size_t)255; return p;
    };
    float*    hA  = (float*)   carve((size_t)NN * 512 * 4);  // transformed features
    float*    hB  = (float*)   carve((size_t)NN * 512 * 4);  // aggregate / layer out
    float*    as_ = (float*)   carve((size_t)NN * 4 * 4);
    float*    ad_ = (float*)   carve((size_t)NN * 4 * 4);
    float*    ee_ = (float*)   carve((size_t)Etot * 4 * 4);  // edge logits
    unsigned* m_  = (unsigned*)carve((size_t)NN * 4 * 4);
    float*    z_  = (float*)   carve((size_t)NN * 4 * 4);
    (void)ws_size; (void)n_in; (void)in_sizes; (void)out_size;

    const dim3 blk(256);

    auto gemm = [&](const float* A, const float* W, const float* bias, float* C,
                    int K, int Nout, int flags) {
        int ngrps = Nout >> 6;
        int wpn   = (ngrps < 8) ? ngrps : 8;
        int mpb   = 8 / wpn;
        dim3 grid((ngrps + wpn - 1) / wpn, (M16 + mpb - 1) / mpb);
        gemm_wmma_f32<<<grid, blk, 0, stream>>>(A, W, bias, C, M16, K, Nout, flags);
    };

    auto gat_layer = [&](const float* in_feat, int Cin, const float* W,
                         const float* a_s, const float* a_d, const float* bias,
                         int heads, float* acc_out, int relu) {
        const int HC = heads * CH;
        // 1) h = in @ W^T  -> hA  (no bias, no relu: bias applies post-aggregation)
        gemm(in_feat, W, nullptr, hA, Cin, HC, 0);
        // 2) attention logits per node
        {
            int cnt = NN * heads;
            alpha_kernel<<<dim3((cnt + 255) / 256), blk, 0, stream>>>(
                hA, a_s, a_d, as_, ad_, heads);
        }
        // 3) init reductions (enc ordering: 0u < enc(any finite float))
        hipMemsetAsync(m_, 0, (size_t)NN * heads * 4, stream);
        hipMemsetAsync(z_, 0, (size_t)NN * heads * 4, stream);
        hipMemsetAsync(acc_out, 0, (size_t)NN * HC * 4, stream);
        // 4) segment max over destinations
        edge_max_kernel<<<dim3((Etot + 255) / 256), blk, 0, stream>>>(
            ei, as_, ad_, ee_, m_, heads, Etot);
        // 5) fused exp + denominator + weighted scatter (one wave per edge)
        edge_agg_kernel<<<dim3((Etot * 32 + 255) / 256), blk, 0, stream>>>(
            ei, ee_, m_, z_, hA, acc_out, heads, Etot);
        // 6) normalize + bias (+relu)
        {
            long long cnt = (long long)NN * HC;
            normalize_kernel<<<dim3((unsigned)((cnt + 255) / 256)), blk, 0, stream>>>(
                acc_out, z_, bias, acc_out, heads, relu);
        }
    };

    // Layer 0: x[N,512] -> hB[N,512], relu
    gat_layer(x,  512, W0, a_src0, a_dst0, b0, 4, hB, 1);
    // Layer 1: hB[N,512] -> hB[N,512], relu (hB free as gemm input once hA is built)
    gat_layer(hB, 512, W1, a_src1, a_dst1, b1, 4, hB, 1);
    // Layer 2: hB[N,512] -> h final in d_out[0 : N*128], heads=1, no relu
    gat_layer(hB, 512, W2, a_src2, a_dst2, b2, 1, out, 0);

    // Output heads: x_vision = relu(h @ Wv^T + bv), x_text = relu(h @ Wt^T + bt)
    gemm(out, Wv, bv, out + (size_t)NN * CH,     CH, CH, 3);
    gemm(out, Wt, bt, out + (size_t)NN * CH * 2, CH, CH, 3);
}